// StructualSimilarity_1116691497094
// MI455X (gfx1250) — compile-verified
//
#include <hip/hip_runtime.h>
#include <hip/hip_bf16.h>
#include <stdint.h>

// Problem constants
#define HN   4096          // image H == W
#define OUTN 4090          // valid-conv output size (H - 6)
#define TX   256           // output columns per block (1 per thread)
#define TY   128           // output rows per block
#define CIN  262           // input columns needed per block (TX + 6)
#define CPAD 264           // padded LDS row stride (66 quads)
#define NBUF 3             // triple-buffered rows: 2-deep async pipeline

// ---------------------------------------------------------------------------
// CDNA5 async global->LDS helpers (gfx1250). ASYNCcnt-tracked DMA into LDS.
// GVS addressing: mem = SGPR64_base + VGPR_i32_offset.
// vdst carries the 32-bit LDS byte address (low 32 bits of a generic shared ptr).
// ---------------------------------------------------------------------------
__device__ __forceinline__ uint32_t lds_addr_of(const void* p) {
  // LDS aperture: addr[63:32] = SHARED_BASE, addr[31:0] = LDS byte offset.
  return (uint32_t)(uintptr_t)p;
}

__device__ __forceinline__ void async_ld_b128(const float* gbase, int byte_off,
                                              uint32_t lds_addr) {
  asm volatile("global_load_async_to_lds_b128 %0, %1, %2"
               :: "v"(lds_addr), "v"(byte_off), "s"(gbase)
               : "memory");
}

// Wait until at most `n` async ops remain outstanding (in-order retirement:
// each staged row = 2 async instructions per issuing wave, so n=2 means
// "all rows except the newest have landed in LDS").
#if __has_builtin(__builtin_amdgcn_s_wait_asynccnt)
#define WAIT_ASYNC_LE(n) __builtin_amdgcn_s_wait_asynccnt(n)
#else
#define WAIT_ASYNC_LE(n) asm volatile("s_wait_asynccnt %0" :: "i"(n) : "memory")
#endif

// ---------------------------------------------------------------------------
// Main kernel: fused separable 7x7 box sums + SSIM math + per-block reduction.
// One thread per output column; slide down TY+6 input rows.
// ---------------------------------------------------------------------------
__global__ __launch_bounds__(256) void ssim_main_kernel(
    const float* __restrict__ warped,
    const float* __restrict__ fimg,
    const float* __restrict__ disp,
    const float* __restrict__ kern,
    float* __restrict__ partial)
{
  __shared__ float raw_w[NBUF][CPAD]; // staged warped rows (async B128 DMA target)
  __shared__ float raw_f[NBUF][CPAD]; // staged fixed rows (async B128 DMA target)
  __shared__ float p_iv[NBUF][CPAD];  // invalid flag (0/1)
  __shared__ float hring[7][6][TX];   // 7-row ring of horizontal 7-sums
  __shared__ float red_v[TX];
  __shared__ float red_c[TX];

  const int tid = threadIdx.x;
  const int x0  = blockIdx.x * TX;
  const int y0  = blockIdx.y * TY;
  const float k0 = kern[0];          // uniform kernel weight (1/49^2)

  // b128 staging: threads 0..65 own quad q -> elements 4q..4q+3 of w and f.
  // Clamp the quad start so the address stays 16B aligned and in-image;
  // clamped values only feed discarded out-of-range outputs.
  const bool do_quad = (tid < (CPAD / 4));          // 66 quads
  int gxq = x0 + 4 * tid;
  gxq = gxq < (HN - 4) ? gxq : (HN - 4);

  // disp staging: threads 0..130 own pixel pair 2t, 2t+1 (float4 load).
  const bool do_pair = (tid < ((CIN + 1) / 2));     // 131 pairs
  int gxp = x0 + 2 * tid;
  gxp = gxp < (HN - 2) ? gxp : (HN - 2);

  auto stage_async = [&](int i, int b) {
    int gy = y0 + i; gy = gy < (HN - 1) ? gy : (HN - 1);
    if (do_quad) {
      const int off = (gy * HN + gxq) * 4;          // byte offset, < 64MB
      const int e   = 4 * tid;
      async_ld_b128(warped, off, lds_addr_of(&raw_w[b][e]));
      async_ld_b128(fimg,   off, lds_addr_of(&raw_f[b][e]));
    }
  };
  auto load_disp = [&](int i) -> float4 {
    int gy = y0 + i; gy = gy < (HN - 1) ? gy : (HN - 1);
    if (do_pair) return *(const float4*)(disp + (size_t)(gy * HN + gxp) * 2);
    return make_float4(0.f, 0.f, 0.f, 0.f);
  };

  float Sw = 0.f, Sf = 0.f, Sww = 0.f, Sff = 0.f, Swf = 0.f, Siv = 0.f;
  float acc_v = 0.f, acc_c = 0.f;
  int slot = 6;                       // rotating ring slot  == i % 7
  int b    = NBUF - 1;                // rotating row buffer == i % 3
  int b2   = 1;                       // rotating prefetch buffer == (i+2) % 3

  // prologue: 2 rows of DMA in flight, disp row 0 pending in registers
  float4 dpend = load_disp(0);
  stage_async(0, 0);
  stage_async(1, 1);

  for (int i = 0; i < TY + 6; ++i) {
    b    = (b    == NBUF - 1) ? 0 : b + 1;
    b2   = (b2   == NBUF - 1) ? 0 : b2 + 1;
    slot = (slot == 6)        ? 0 : slot + 1;

    // commit invalid flags for row i (loaded one iteration ago)
    if (do_pair) {
      const int e = 2 * tid;
      const float iv0 = (fabsf(dpend.x) > 1.0f || fabsf(dpend.y) > 1.0f) ? 1.0f : 0.0f;
      const float iv1 = (fabsf(dpend.z) > 1.0f || fabsf(dpend.w) > 1.0f) ? 1.0f : 0.0f;
      *(float2*)&p_iv[b][e] = make_float2(iv0, iv1);
    }

    WAIT_ASYNC_LE(2);                 // rows <= i are in LDS; row i+1 may fly
    __syncthreads();                  // all waves' raw + iv of row i visible

    // keep the pipeline 2 rows deep (buffer b2 free: its last readers were
    // iteration i-1's hsum, finished before this barrier)
    if (i + 2 < TY + 6) stage_async(i + 2, b2);
    if (i + 1 < TY + 6) dpend = load_disp(i + 1);

    // horizontal 7-sums at column tid; products recomputed in registers
    float hw = 0.f, hf = 0.f, hww = 0.f, hff = 0.f, hwf = 0.f, hiv = 0.f;
#pragma unroll
    for (int d = 0; d < 7; ++d) {
      const float a = raw_w[b][tid + d];
      const float c = raw_f[b][tid + d];
      hw += a;
      hf += c;
      hww = fmaf(a, a, hww);
      hff = fmaf(c, c, hff);
      hwf = fmaf(a, c, hwf);
      hiv += p_iv[b][tid + d];
    }

    // vertical sliding window via 7-deep ring (thread-private slots)
    if (i >= 7) {
      Sw  -= hring[slot][0][tid];
      Sf  -= hring[slot][1][tid];
      Sww -= hring[slot][2][tid];
      Sff -= hring[slot][3][tid];
      Swf -= hring[slot][4][tid];
      Siv -= hring[slot][5][tid];
    }
    Sw += hw; Sf += hf; Sww += hww; Sff += hff; Swf += hwf; Siv += hiv;
    hring[slot][0][tid] = hw;
    hring[slot][1][tid] = hf;
    hring[slot][2][tid] = hww;
    hring[slot][3][tid] = hff;
    hring[slot][4][tid] = hwf;
    hring[slot][5][tid] = hiv;

    if (i >= 6) {
      const int oy = y0 + i - 6;
      const int ox = x0 + tid;
      // Siv is an exact small integer in f32 -> ==0 matches conv(inv)==0.
      const bool keep = (Siv == 0.0f) && (oy < OUTN) && (ox < OUTN);
      if (keep) {
        const float mw  = Sw  * k0, mf  = Sf  * k0;
        const float mww = Sww * k0, mff = Sff * k0, mwf = Swf * k0;
        const float var_w = mww - mw * mw;
        const float var_f = mff - mf * mf;
        const float cov   = mwf - mw * mf;
        const float sp = sqrtf((var_f + 1e-10f) * (var_w + 1e-10f)); // std_f*std_w
        const float num = (2.0f * mf * mw + 1e-5f) *
                          (2.0f * sp + 1e-5f) *
                          (cov + 1e-5f);
        const float den = (mf * mf + mw * mw + 1e-5f) *
                          (var_f + var_w + 1e-5f) *
                          (sp + 1e-5f);
        acc_v -= num / den;           // val = -sim
        acc_c += 1.0f;
      }
    }
  }

  // deterministic per-block tree reduction
  red_v[tid] = acc_v;
  red_c[tid] = acc_c;
  __syncthreads();
  for (int s = TX / 2; s > 0; s >>= 1) {
    if (tid < s) { red_v[tid] += red_v[tid + s]; red_c[tid] += red_c[tid + s]; }
    __syncthreads();
  }
  if (tid == 0) {
    const int bid = blockIdx.y * gridDim.x + blockIdx.x;
    partial[2 * bid]     = red_v[0];
    partial[2 * bid + 1] = red_c[0];
  }
}

// ---------------------------------------------------------------------------
// Final fixed-order reduction of per-block partials -> scalar mean
// ---------------------------------------------------------------------------
__global__ __launch_bounds__(256) void ssim_final_kernel(
    const float* __restrict__ partial, int n, float* __restrict__ out)
{
  __shared__ double sv[256];
  __shared__ double sc[256];
  const int tid = threadIdx.x;
  double v = 0.0, c = 0.0;
  for (int i = tid; i < n; i += 256) {
    v += (double)partial[2 * i];
    c += (double)partial[2 * i + 1];
  }
  sv[tid] = v; sc[tid] = c;
  __syncthreads();
  for (int s = 128; s > 0; s >>= 1) {
    if (tid < s) { sv[tid] += sv[tid + s]; sc[tid] += sc[tid + s]; }
    __syncthreads();
  }
  if (tid == 0) {
    const double cnt = sc[0] > 1.0 ? sc[0] : 1.0;
    out[0] = (float)(sv[0] / cnt);
  }
}

// ---------------------------------------------------------------------------
// Launch: inputs per setup_inputs() order:
//   d_in[0]=warped (4096^2 f32), d_in[1]=moving (unused), d_in[2]=fixed,
//   d_in[3]=displacement (4096^2 x2 f32), d_in[4]=kernel (7x7 f32)
// d_out: 1 f32 scalar. d_ws: >= 2*512 f32 partials.
// ---------------------------------------------------------------------------
extern "C" void kernel_launch(void* const* d_in, const int* in_sizes, int n_in,
                              void* d_out, int out_size, void* d_ws, size_t ws_size,
                              hipStream_t stream) {
  (void)in_sizes; (void)n_in; (void)out_size; (void)ws_size;
  const float* warped = (const float*)d_in[0];
  const float* fimg   = (const float*)d_in[2];
  const float* disp   = (const float*)d_in[3];
  const float* kern   = (const float*)d_in[4];
  float* partial = (float*)d_ws;

  dim3 grid((HN + TX - 1) / TX, (HN + TY - 1) / TY);  // 16 x 32 = 512 blocks
  ssim_main_kernel<<<grid, TX, 0, stream>>>(warped, fimg, disp, kern, partial);

  const int nblocks = (int)(grid.x * grid.y);
  ssim_final_kernel<<<1, 256, 0, stream>>>(partial, nblocks, (float*)d_out);
}